// GCN_87720412054223
// MI455X (gfx1250) — compile-verified
//
#include <hip/hip_runtime.h>

typedef __attribute__((ext_vector_type(2))) float v2f;
typedef __attribute__((ext_vector_type(8))) float v8f;

// ---------------------------------------------------------------------------
// K1: init degree accumulator to 1.0 (the self-loop)
__global__ __launch_bounds__(256) void k_init_deg(float* __restrict__ deg, int n) {
  int i = blockIdx.x * blockDim.x + threadIdx.x;
  if (i < n) deg[i] = 1.0f;
}

// K2: degree scatter, vectorized (e % 4 == 0 path)
__global__ __launch_bounds__(256) void k_deg_v4(const int* __restrict__ dst,
                                                float* __restrict__ deg, int nvec) {
  int i = blockIdx.x * blockDim.x + threadIdx.x;
  if (i < nvec) {
    int4 d = ((const int4*)dst)[i];
    atomicAdd(&deg[d.x], 1.0f);
    atomicAdd(&deg[d.y], 1.0f);
    atomicAdd(&deg[d.z], 1.0f);
    atomicAdd(&deg[d.w], 1.0f);
  }
}
__global__ __launch_bounds__(256) void k_deg_s(const int* __restrict__ dst,
                                               float* __restrict__ deg, int e) {
  int i = blockIdx.x * blockDim.x + threadIdx.x;
  if (i < e) atomicAdd(&deg[dst[i]], 1.0f);
}

// K3: dinv = rsqrt(deg); p = x*dinv; t = p (self-loop contribution)
__global__ __launch_bounds__(256) void k_node1(const float* __restrict__ x,
                                               const float* __restrict__ deg,
                                               float* __restrict__ dinv,
                                               float* __restrict__ p,
                                               float* __restrict__ t, int n) {
  int i = blockIdx.x * blockDim.x + threadIdx.x;
  if (i < n) {
    float dv = rsqrtf(deg[i]);
    dinv[i] = dv;
    float pv = x[i] * dv;
    p[i] = pv;
    t[i] = pv;
  }
}

// K4/K6: scalar edge scatter acc[dst] += val[src], 4 edges/thread
__global__ __launch_bounds__(256) void k_scatter_v4(const int* __restrict__ src,
                                                    const int* __restrict__ dst,
                                                    const float* __restrict__ val,
                                                    float* __restrict__ acc, int nvec) {
  int i = blockIdx.x * blockDim.x + threadIdx.x;
  if (i < nvec) {
    int4 s = ((const int4*)src)[i];
    int4 d = ((const int4*)dst)[i];
    atomicAdd(&acc[d.x], val[s.x]);
    atomicAdd(&acc[d.y], val[s.y]);
    atomicAdd(&acc[d.z], val[s.z]);
    atomicAdd(&acc[d.w], val[s.w]);
  }
}
__global__ __launch_bounds__(256) void k_scatter_s(const int* __restrict__ src,
                                                   const int* __restrict__ dst,
                                                   const float* __restrict__ val,
                                                   float* __restrict__ acc, int e) {
  int i = blockIdx.x * blockDim.x + threadIdx.x;
  if (i < e) atomicAdd(&acc[dst[i]], val[src[i]]);
}

// ---------------------------------------------------------------------------
// xor-butterfly add via ds_swizzle_b32 immediate (group-of-32 mode:
// and=0x1f, or=0, xor=mask -> SWAPX<mask> pattern; no LDS memory touched).
template <int IMM>
__device__ __forceinline__ float swz_add(float v) {
#if defined(__AMDGCN__)
  int s = __builtin_amdgcn_ds_swizzle(__float_as_int(v), IMM);
  return v + __int_as_float(s);
#else
  return v;  // host parse only
#endif
}

// K5: per-16-node tile, via v_wmma_f32_16x16x4_f32.
//   a[m] = dinv[m]*t[m]
//   H = A x B + C : A[m][0]=a[m], B[0][n]=W1[n], C[m][n]=b1[n]   (rank-1, fp32)
//   z[m] = sum_n relu(H[m][n])*W2[n]   (16-lane xor-swizzle reduction)
//   q[m] = u[m] = z[m]*dinv[m]
__global__ __launch_bounds__(256) void k_mid(const float* __restrict__ tacc,
                                             const float* __restrict__ dinv,
                                             const float* __restrict__ W1,
                                             const float* __restrict__ b1,
                                             const float* __restrict__ W2,
                                             float* __restrict__ q,
                                             float* __restrict__ u, int n) {
  const int lane   = threadIdx.x & 31;
  const int lane16 = lane & 15;
  const bool hi    = lane >= 16;
  const long long wave = (long long)blockIdx.x * (blockDim.x >> 5) + (threadIdx.x >> 5);
  const long long base = wave * 16;
  if (base >= n) return;                       // wave-uniform: EXEC stays full

  float a = 0.0f;
  if (!hi) {
    long long idx = base + lane16;
    if (idx < n) a = tacc[idx] * dinv[idx];
  }
  const float w1v = W1[lane16];
  const float b1v = b1[lane16];
  const float w2v = W2[lane16];

#if __has_builtin(__builtin_amdgcn_wmma_f32_16x16x4_f32)
  // A 16x4 f32: lanes 0-15 -> {K0,K1}, lanes 16-31 -> {K2,K3}; only K0 nonzero
  v2f A; A.x = a;                  A.y = 0.0f;   // a==0 on hi lanes
  // B 4x16 f32: lanes 0-15 -> {K0,K1}; only row K0 = W1
  v2f B; B.x = hi ? 0.0f : w1v;    B.y = 0.0f;
  v8f C;
  #pragma unroll
  for (int r = 0; r < 8; ++r) C[r] = b1v;       // C[m][n] = b1[n]
  v8f D = __builtin_amdgcn_wmma_f32_16x16x4_f32(
      false, A, false, B, (short)0, C, false, false);
#else
  // fallback: rank-1 outer product directly in D layout via broadcasts
  v8f D;
  #pragma unroll
  for (int r = 0; r < 8; ++r) {
    float am = __shfl(a, r + (hi ? 8 : 0), 32);
    D[r] = fmaf(am, w1v, b1v);
  }
#endif

  // D layout: VGPR r, lanes 0-15 -> H[m=r][n=lane]; lanes 16-31 -> H[m=r+8][n=lane-16]
  float z[8];
  #pragma unroll
  for (int r = 0; r < 8; ++r) {
    float h = D[r];
    h = h > 0.0f ? h : 0.0f;                    // relu
    float v = h * w2v;
    v = swz_add<0x041F>(v);                     // xor 1 (stays inside 16-lane half)
    v = swz_add<0x081F>(v);                     // xor 2
    v = swz_add<0x101F>(v);                     // xor 4
    v = swz_add<0x201F>(v);                     // xor 8
    z[r] = v;                                   // lanes<16: z[m=r]; lanes>=16: z[m=r+8]
  }

  if (lane == 0 || lane == 16) {
    const int off = hi ? 8 : 0;
    #pragma unroll
    for (int r = 0; r < 8; ++r) {
      long long idx = base + off + r;
      if (idx < n) {
        float qq = z[r] * dinv[idx];
        q[idx] = qq;
        u[idx] = qq;                            // self-loop init for layer-2 scatter
      }
    }
  }
}

// K7: out = dinv*u + b2
__global__ __launch_bounds__(256) void k_out(const float* __restrict__ dinv,
                                             const float* __restrict__ u,
                                             const float* __restrict__ b2,
                                             float* __restrict__ out, int n) {
  int i = blockIdx.x * blockDim.x + threadIdx.x;
  if (i < n) out[i] = fmaf(dinv[i], u[i], b2[0]);
}

// ---------------------------------------------------------------------------
extern "C" void kernel_launch(void* const* d_in, const int* in_sizes, int n_in,
                              void* d_out, int out_size, void* d_ws, size_t ws_size,
                              hipStream_t stream) {
  const float* x  = (const float*)d_in[0];
  const float* W1 = (const float*)d_in[1];
  const float* b1 = (const float*)d_in[2];
  const float* W2 = (const float*)d_in[3];
  const float* b2 = (const float*)d_in[4];
  const int*   ei = (const int*)d_in[5];

  const int n = in_sizes[0];        // 500,000 nodes
  const int e = in_sizes[5] / 2;    // 5,000,000 edges
  const int* src = ei;
  const int* dst = ei + e;

  float* out  = (float*)d_out;      // also used as degree accumulator early on
  float* dinv = (float*)d_ws;       // [n]
  float* p    = dinv + n;           // [n]  (reused as q)
  float* t    = p + n;              // [n]
  float* u    = t + n;              // [n]
  float* q    = p;

  const int B  = 256;
  const int gN = (n + B - 1) / B;
  const int gM = (n + 127) / 128;   // 8 waves/block * 16 nodes/wave

  // 1) degree (into d_out): self loop + incoming edges
  k_init_deg<<<gN, B, 0, stream>>>(out, n);
  if ((e & 3) == 0) {
    int nvec = e >> 2;
    int gE = (nvec + B - 1) / B;
    k_deg_v4<<<gE, B, 0, stream>>>(dst, out, nvec);
  } else {
    int gE = (e + B - 1) / B;
    k_deg_s<<<gE, B, 0, stream>>>(dst, out, e);
  }

  // 2) dinv, p = x*dinv, t = p
  k_node1<<<gN, B, 0, stream>>>(x, out, dinv, p, t, n);

  // 3) layer-1 scalar scatter: t[dst] += p[src]
  if ((e & 3) == 0) {
    int nvec = e >> 2;
    int gE = (nvec + B - 1) / B;
    k_scatter_v4<<<gE, B, 0, stream>>>(src, dst, p, t, nvec);
  } else {
    int gE = (e + B - 1) / B;
    k_scatter_s<<<gE, B, 0, stream>>>(src, dst, p, t, e);
  }

  // 4) WMMA middle: q = u = dinv * (W2 . relu(outer(dinv*t, W1) + b1))
  k_mid<<<gM, B, 0, stream>>>(t, dinv, W1, b1, W2, q, u, n);

  // 5) layer-2 scalar scatter: u[dst] += q[src]
  if ((e & 3) == 0) {
    int nvec = e >> 2;
    int gE = (nvec + B - 1) / B;
    k_scatter_v4<<<gE, B, 0, stream>>>(src, dst, q, u, nvec);
  } else {
    int gE = (e + B - 1) / B;
    k_scatter_s<<<gE, B, 0, stream>>>(src, dst, q, u, e);
  }

  // 6) out = dinv*u + b2
  k_out<<<gN, B, 0, stream>>>(dinv, u, b2, out, n);
}